// SelectiveSSM_30691836297503
// MI455X (gfx1250) — compile-verified
//
#include <hip/hip_runtime.h>
#include <math.h>

// Problem constants (match reference).
namespace {
constexpr int kB = 2, kL = 2048, kD = 768, kN = 16, kR = 48;
constexpr int kBL = kB * kL;  // 4096 rows in the (B*L) x D activation matrices
}

typedef float v2f __attribute__((ext_vector_type(2)));
typedef float v8f __attribute__((ext_vector_type(8)));

// One V_WMMA_F32_16X16X4_F32 step (full f32, matches reference numerics).
__device__ __forceinline__ v8f wmma4(v2f a, v2f b, v8f c) {
  return __builtin_amdgcn_wmma_f32_16x16x4_f32(false, a, false, b, (short)0, c,
                                               false, false);
}

// ------- Kernel 0: transpose weights so WMMA B-operands load as b64 -------
// WT_BC[32][768] <- W_BC[768][32]; WT_dt[48][768] <- W_dt[768][48];
// WT_dtp[768][48] <- W_dtp[48][768].
__global__ __launch_bounds__(256) void k_prep_weights(
    const float* __restrict__ W_BC, const float* __restrict__ W_dt,
    const float* __restrict__ W_dtp, float* __restrict__ WT_BC,
    float* __restrict__ WT_dt, float* __restrict__ WT_dtp) {
  int idx = blockIdx.x * 256 + threadIdx.x;
  const int t1 = 32 * kD, t2 = kR * kD, t3 = kD * kR;
  if (idx < t1) {
    int n = idx / kD, k = idx % kD;
    WT_BC[idx] = W_BC[k * 32 + n];
  } else if (idx < t1 + t2) {
    int j = idx - t1;
    int n = j / kD, k = j % kD;
    WT_dt[j] = W_dt[k * kR + n];
  } else if (idx < t1 + t2 + t3) {
    int j = idx - t1 - t2;
    int n = j / kR, k = j % kR;
    WT_dtp[j] = W_dtp[k * kD + n];
  }
}

// ---------------- Kernel 1: depthwise conv(3, SAME) + SiLU -> u ----------------
__global__ __launch_bounds__(256) void k_conv_silu(const float* __restrict__ x,
                                                   const float* __restrict__ w,
                                                   float* __restrict__ u) {
  int idx = blockIdx.x * 256 + threadIdx.x;
  if (idx >= kBL * kD) return;
  int d = idx % kD;
  int l = (idx / kD) % kL;
  float s = w[kD + d] * x[idx];                      // k[1]*x[l]
  if (l > 0)      s += w[d] * x[idx - kD];           // k[0]*x[l-1]
  if (l < kL - 1) s += w[2 * kD + d] * x[idx + kD];  // k[2]*x[l+1]
  u[idx] = s / (1.f + __expf(-s));                   // SiLU
}

// ------- Kernel 2: BC = u@W_BC + b_BC ; dtlow = u@W_dt + b_dt -------
// One wave per 16-row tile computes all 5 output col tiles with 5 accumulators:
// per K-step 6 contiguous b64 loads feed 5 independent WMMAs.
__global__ __launch_bounds__(256) void k_gemm_bc_dt(
    const float* __restrict__ u, const float* __restrict__ WT_BC,
    const float* __restrict__ b_BC, const float* __restrict__ WT_dt,
    const float* __restrict__ b_dt, float* __restrict__ BC,
    float* __restrict__ dtlow) {
  int wave = (blockIdx.x * 256 + threadIdx.x) >> 5;
  if (wave >= kBL / 16) return;
  const int rt = wave;
  const int lane = threadIdx.x & 31;
  const int mn = lane & 15;         // A row / B col within tile
  const int kh = (lane >> 4) << 1;  // K sub-offset: lanes 0-15 -> 0, 16-31 -> 2

  const float* ap  = u + ((size_t)rt * 16 + mn) * kD + kh;
  const float* bp0 = WT_BC + (size_t)(mn) * kD + kh;        // BC cols 0-15
  const float* bp1 = WT_BC + (size_t)(16 + mn) * kD + kh;   // BC cols 16-31
  const float* bp2 = WT_dt + (size_t)(mn) * kD + kh;        // dt cols 0-15
  const float* bp3 = WT_dt + (size_t)(16 + mn) * kD + kh;   // dt cols 16-31
  const float* bp4 = WT_dt + (size_t)(32 + mn) * kD + kh;   // dt cols 32-47

  v8f a0 = {}, a1 = {}, a2 = {}, a3 = {}, a4 = {};
#pragma unroll 4
  for (int kb = 0; kb < kD; kb += 4) {
    v2f a = *(const v2f*)(ap + kb);
    v2f b0 = *(const v2f*)(bp0 + kb);
    v2f b1 = *(const v2f*)(bp1 + kb);
    v2f b2 = *(const v2f*)(bp2 + kb);
    v2f b3 = *(const v2f*)(bp3 + kb);
    v2f b4 = *(const v2f*)(bp4 + kb);
    a0 = wmma4(a, b0, a0);
    a1 = wmma4(a, b1, a1);
    a2 = wmma4(a, b2, a2);
    a3 = wmma4(a, b3, a3);
    a4 = wmma4(a, b4, a4);
  }

  const int mo = (lane >> 4) * 8;  // C/D: VGPR v -> row v (lanes 0-15) / v+8
  float bb0 = b_BC[mn], bb1 = b_BC[16 + mn];
  float bd0 = b_dt[mn], bd1 = b_dt[16 + mn], bd2 = b_dt[32 + mn];
#pragma unroll
  for (int v = 0; v < 8; ++v) {
    size_t row = (size_t)rt * 16 + mo + v;
    BC[row * 32 + mn]         = a0[v] + bb0;
    BC[row * 32 + 16 + mn]    = a1[v] + bb1;
    dtlow[row * kR + mn]      = a2[v] + bd0;
    dtlow[row * kR + 16 + mn] = a3[v] + bd1;
    dtlow[row * kR + 32 + mn] = a4[v] + bd2;
  }
}

// ------- Kernel 3: dt = softplus(dtlow @ W_dtp + b_dtp), K=48 -------
// A tile (16x48) preloaded into 12 v2f regs; each wave sweeps 4 col tiles with
// 12 back-to-back WMMAs each (all operands register/L2 resident).
__global__ __launch_bounds__(256) void k_gemm_dtproj(
    const float* __restrict__ dtlow, const float* __restrict__ WT_dtp,
    const float* __restrict__ b_dtp, float* __restrict__ dt) {
  int wave = (blockIdx.x * 256 + threadIdx.x) >> 5;
  const int groups = kD / 64;  // 12 groups of 4 col tiles
  int rt = wave / groups, g = wave % groups;
  if (rt >= kBL / 16) return;
  const int lane = threadIdx.x & 31;
  const int mn = lane & 15;
  const int kh = (lane >> 4) << 1;

  const float* ap = dtlow + ((size_t)rt * 16 + mn) * kR + kh;
  v2f a[12];
#pragma unroll
  for (int i = 0; i < 12; ++i) a[i] = *(const v2f*)(ap + 4 * i);

  const int mo = (lane >> 4) * 8;
#pragma unroll
  for (int t = 0; t < 4; ++t) {
    const int col0 = g * 64 + t * 16;
    const float* bp = WT_dtp + (size_t)(col0 + mn) * kR + kh;
    v8f acc = {};
#pragma unroll
    for (int i = 0; i < 12; ++i) {
      v2f b = *(const v2f*)(bp + 4 * i);
      acc = wmma4(a[i], b, acc);
    }
    float bv = b_dtp[col0 + mn];
#pragma unroll
    for (int v = 0; v < 8; ++v) {
      float s = acc[v] + bv;
      s = (s > 20.f) ? s : log1pf(__expf(s));  // softplus
      dt[((size_t)rt * 16 + mo + v) * kD + col0 + mn] = s;
    }
  }
}

// ---------------- Kernel 4: selective scan + output ----------------
// One thread per (b, d, n); each 16-lane half-wave owns one channel's 16 states.
// x/dt loaded once per channel per 16-step chunk, broadcast via shuffles;
// y contribution reduced across n with an intra-group butterfly.
__global__ __launch_bounds__(256) void k_scan(
    const float* __restrict__ x, const float* __restrict__ dt,
    const float* __restrict__ BC, const float* __restrict__ A_log,
    const float* __restrict__ Dcoeff, float* __restrict__ y) {
  int b = blockIdx.x / (kD / 16);
  int dBase = (blockIdx.x % (kD / 16)) * 16;
  int t = threadIdx.x;
  int d = dBase + (t >> 4);
  int n = t & 15;
  int half = t & 16;  // which 16-lane group inside the wave32
  float Ac = -__expf(A_log[d * kN + n]);
  float Dd = Dcoeff[d];
  const float* xb  = x  + (size_t)b * kL * kD + d;
  const float* dtb = dt + (size_t)b * kL * kD + d;
  const float* bcb = BC + (size_t)b * kL * (2 * kN);
  float* yb = y + (size_t)b * kL * kD + d;
  float h = 0.f;
  for (int lc = 0; lc < kL; lc += 16) {
    float xc = xb[(size_t)(lc + n) * kD];  // lane n prefetches l = lc+n
    float dc = dtb[(size_t)(lc + n) * kD];
    for (int j = 0; j < 16; ++j) {
      int l = lc + j;
      float xl  = __shfl(xc, half | j, 32);
      float dtl = __shfl(dc, half | j, 32);
      float Bcv = bcb[l * (2 * kN) + n];
      float Ccv = bcb[l * (2 * kN) + kN + n];
      h = __expf(dtl * Ac) * h + Bcv * xl * dtl;  // h_t = dA*h + dB
      float c = Ccv * h;
      c += __shfl_xor(c, 8, 32);
      c += __shfl_xor(c, 4, 32);
      c += __shfl_xor(c, 2, 32);
      c += __shfl_xor(c, 1, 32);
      if (n == 0) yb[(size_t)l * kD] = c + xl * Dd;
    }
  }
}

extern "C" void kernel_launch(void* const* d_in, const int* in_sizes, int n_in,
                              void* d_out, int out_size, void* d_ws,
                              size_t ws_size, hipStream_t stream) {
  (void)in_sizes; (void)n_in; (void)out_size; (void)ws_size;
  const float* x     = (const float*)d_in[0];
  const float* ck    = (const float*)d_in[1];
  const float* A_log = (const float*)d_in[2];
  const float* W_BC  = (const float*)d_in[3];
  const float* b_BC  = (const float*)d_in[4];
  const float* Dco   = (const float*)d_in[5];
  const float* W_dt  = (const float*)d_in[6];
  const float* b_dt  = (const float*)d_in[7];
  const float* W_dtp = (const float*)d_in[8];
  const float* b_dtp = (const float*)d_in[9];
  float* y  = (float*)d_out;
  float* ws = (float*)d_ws;

  // Workspace layout (floats). u/dt alias (u dead after k_gemm_bc_dt).
  float* u      = ws;                               // [kBL, kD]
  float* BCbuf  = u + (size_t)kBL * kD;             // [kBL, 32]
  float* dtlow  = BCbuf + (size_t)kBL * 2 * kN;     // [kBL, 48]
  float* WT_BC  = dtlow + (size_t)kBL * kR;         // [32, 768]
  float* WT_dt  = WT_BC + (size_t)32 * kD;          // [48, 768]
  float* WT_dtp = WT_dt + (size_t)kR * kD;          // [768, 48]
  float* dtbuf  = u;                                // [kBL, kD] (reuse)

  const int prep_total = 32 * kD + kR * kD + kD * kR;  // 98304
  k_prep_weights<<<(prep_total + 255) / 256, 256, 0, stream>>>(
      W_BC, W_dt, W_dtp, WT_BC, WT_dt, WT_dtp);
  k_conv_silu<<<(kBL * kD + 255) / 256, 256, 0, stream>>>(x, ck, u);
  // 256 row tiles, one wave each (5 col tiles per wave) -> 32 blocks.
  k_gemm_bc_dt<<<(kBL / 16) * 32 / 256, 256, 0, stream>>>(
      u, WT_BC, b_BC, WT_dt, b_dt, BCbuf, dtlow);
  // 256 row tiles * 12 groups -> 3072 waves -> 384 blocks.
  k_gemm_dtproj<<<(kBL / 16) * (kD / 64) * 32 / 256, 256, 0, stream>>>(
      dtlow, WT_dtp, b_dtp, dtbuf);
  k_scan<<<kB * (kD / 16), 256, 0, stream>>>(x, dtbuf, BCbuf, A_log, Dco, y);
}